// AEQ_SP_26001732010264
// MI455X (gfx1250) — compile-verified
//
#include <hip/hip_runtime.h>
#include <stdint.h>

// CMA adaptive equalizer: serial recurrence, latency-bound.
// One wave32, tap-per-lane, register sliding window, async global->LDS
// double-buffered streaming of y, and V_WMMA_F32_16X16X4_F32 used as the
// cross-lane reducer for the per-iteration complex dot product.

#define EQLEN 31
#define CH 2048            // complex elements per LDS chunk (16 KB per buffer)
#define CH_LOG 11

#if defined(__AMDGCN__) && __has_builtin(__builtin_amdgcn_global_load_async_to_lds_b128)
#define CMA_ASYNC 1
#else
#define CMA_ASYNC 0
#endif

#if defined(__AMDGCN__) && __has_builtin(__builtin_amdgcn_wmma_f32_16x16x4_f32)
#define CMA_WMMA 1
#else
#define CMA_WMMA 0
#endif

typedef int   cma_v4i __attribute__((ext_vector_type(4)));
typedef float cma_v2f __attribute__((ext_vector_type(2)));
typedef float cma_v8f __attribute__((ext_vector_type(8)));

__device__ __forceinline__ void chunk_issue(float2* dst, const float2* src, int count)
{
    const int lane = (int)threadIdx.x;
#if CMA_ASYNC
    const char* g = (const char*)src;
    char*       l = (char*)dst;
    const int bytes = count * (int)sizeof(float2);
    // 32 lanes x 16B = 512B per async instruction
    for (int off = lane * 16; off < bytes; off += 32 * 16) {
        __builtin_amdgcn_global_load_async_to_lds_b128(
            (__attribute__((address_space(1))) cma_v4i*)(g + off),
            (__attribute__((address_space(3))) cma_v4i*)(l + off),
            0, 0);
    }
#else
    for (int t = lane; t < count; t += 32) dst[t] = src[t];
#endif
}

__device__ __forceinline__ void chunk_wait()
{
#if CMA_ASYNC
#if __has_builtin(__builtin_amdgcn_s_wait_asynccnt)
    __builtin_amdgcn_s_wait_asynccnt(0);
#else
    asm volatile("s_wait_asynccnt 0x0" ::: "memory");
#endif
#endif
    __syncthreads();
}

__global__ __launch_bounds__(32, 1)
void cma_equalize_kernel(const float2* __restrict__ y,
                         const float2* __restrict__ taps0,
                         float2* __restrict__ out,
                         int n, int niter)
{
    __shared__ float2 buf[2][CH];
    const int lane = (int)threadIdx.x;

    // taps: lane m holds taps[m]; lane 31 stays zero (never updated)
    float tr = 0.0f, ti = 0.0f;
    if (lane < EQLEN) { float2 t0 = taps0[lane]; tr = t0.x; ti = t0.y; }

    // initial sliding window: lane m <- y[(k0-1-m) mod n], k0 = 15
    // (negative torch indices wrap, hence the modulo)
    int wi = 14 - lane;
    if (wi < 0) wi += n;
    float2 w0 = y[wi];
    float wr = w0.x, wim = w0.y;

#if CMA_WMMA
    // B selector (4x16, f32): column 0 sums K rows {0,2} (real slots),
    // column 1 sums K rows {1,3} (imag slots). Column j lives split across
    // lane j (K=0,1 in the 2 VGPRs) and lane 16+j (K=2,3).
    cma_v2f bsel;
    bsel.x = ((lane & 15) == 0) ? 1.0f : 0.0f;   // B[0][j] / B[2][j]
    bsel.y = ((lane & 15) == 1) ? 1.0f : 0.0f;   // B[1][j] / B[3][j]
#endif

    const int NCH = (n + CH - 1) / CH;
    chunk_issue(buf[0], y, (n < CH) ? n : CH);
    if (NCH > 1) chunk_issue(buf[1], y + CH, ((n - CH) < CH) ? (n - CH) : CH);
    chunk_wait();

    int  chunks_loaded = (NCH > 1) ? 2 : 1;
    bool pending = false;

    float lr = 1e-3f;
    int next_halve = 20000;

    for (int it = 0; it < niter; ++it) {
        const int k = 15 + 2 * it;

        // lr halves at it = 20000, 40000, ... (matches i%20000==0 && i!=0)
        if (it == next_halve) { lr *= 0.5f; next_halve += 20000; }

        // prefetch chunk c into buf[c&1] once chunk c-2 can no longer be read
        if (!pending && chunks_loaded < NCH && k >= (chunks_loaded - 1) * CH) {
            const int c   = chunks_loaded;
            const int cnt = ((n - c * CH) < CH) ? (n - c * CH) : CH;
            chunk_issue(buf[c & 1], y + (size_t)c * CH, cnt);
            pending = true;
        }
        // block only when the next window value actually needs the new chunk
        if (pending && (k + 1) >= chunks_loaded * CH) {
            chunk_wait();
            pending = false;
            ++chunks_loaded;
        }

        // per-lane complex product: taps[m] * y[k-1-m]
        float pr = tr * wr  - ti * wim;
        float pi = tr * wim + ti * wr;

        float sr, si;
#if CMA_WMMA
        // A (16x4): lane m<16 holds A[m][0..1] = {pr_m, pi_m};
        //           lane 16+m holds A[m][2..3] = {pr_{m+16}, pi_{m+16}}.
        // D = A x Bsel: column 0 = real partials per row, column 1 = imag.
        cma_v2f a; a.x = pr; a.y = pi;
        cma_v8f acc = {};
        acc = __builtin_amdgcn_wmma_f32_16x16x4_f32(
            false, a, false, bsel, (short)0, acc, false, false);
        // lane 0: D[0..7][0], lane 16: D[8..15][0]; lane 1/17 same for imag.
        float t8 = ((acc[0] + acc[1]) + (acc[2] + acc[3])) +
                   ((acc[4] + acc[5]) + (acc[6] + acc[7]));
        float t16 = t8 + __shfl_xor(t8, 16, 32);
        sr = __shfl(t16, 0, 32);     // full real sum (from lane 0)
        si = __shfl(t16, 1, 32);     // full imag sum (from lane 1)
#else
        // wave32 butterfly reduction -> o = (sr, si) broadcast to all lanes
        sr = pr; si = pi;
        #pragma unroll
        for (int off = 16; off >= 1; off >>= 1) {
            sr += __shfl_xor(sr, off, 32);
            si += __shfl_xor(si, off, 32);
        }
#endif

        const float e  = 1.0f - (sr * sr + si * si);  // R - |o|^2
        const float c2 = 2.0f * lr * e;

        // taps[m] += 2*lr*e * (o * conj(y[k-1-m]))  (same y value this lane holds)
        if (lane < EQLEN) {
            tr += c2 * (sr * wr + si * wim);
            ti += c2 * (si * wr - sr * wim);
        }

        if (lane == 0) out[it] = make_float2(sr, si);

        // shift window by SPS=2: new[m] = old[m-2]; lanes 0,1 fetch y[k+1], y[k]
        float nwr = __shfl_up(wr, 2, 32);
        float nwi = __shfl_up(wim, 2, 32);
        if (lane < 2) {
            const int gi = k + 1 - lane;
            const float2 v = buf[(gi >> CH_LOG) & 1][gi & (CH - 1)];
            nwr = v.x; nwi = v.y;
        }
        wr = nwr; wim = nwi;
    }
}

__global__ void zero_f32_kernel(float* __restrict__ p, int count)
{
    int i = blockIdx.x * blockDim.x + threadIdx.x;
    for (; i < count; i += gridDim.x * blockDim.x) p[i] = 0.0f;
}

extern "C" void kernel_launch(void* const* d_in, const int* in_sizes, int n_in,
                              void* d_out, int out_size, void* d_ws, size_t ws_size,
                              hipStream_t stream)
{
    (void)d_ws; (void)ws_size;

    int n = in_sizes[0];
    // Harness may count complex64 as float pairs; taps has 31 complex elements.
    if (n_in > 1 && in_sizes[1] == 2 * EQLEN) n /= 2;

    const float2* y    = (const float2*)d_in[0];
    const float2* taps = (const float2*)d_in[1];

    // niter = len(arange(15, n-31, 2)) = (n-45)/2 for our sizes
    int niter = (n - 45) / 2;
    if (niter < 0) niter = 0;
    long cap = (long)out_size / 2;          // out_size is in floats (complex64 -> 2 floats)
    if ((long)niter > cap) niter = (int)cap;

    int zb = (out_size + 255) / 256;
    if (zb < 1) zb = 1;
    if (zb > 65535) zb = 65535;
    zero_f32_kernel<<<zb, 256, 0, stream>>>((float*)d_out, out_size);

    cma_equalize_kernel<<<1, 32, 0, stream>>>(y, taps, (float2*)d_out, n, niter);
}